// BudgetBisect_90924457656606
// MI455X (gfx1250) — compile-verified
//
#include <hip/hip_runtime.h>

// BudgetBisect on MI455X (gfx1250):
//   - one row (16384 f32 = 64 KB) per 256-thread block, register-resident (64 f32/thread)
//   - row staged once via CDNA5 async-to-LDS pipe (ASYNCcnt), then ds_load_b128 -> VGPRs
//   - per bisection iter: p = v_med3(x - tau, 0, 1)  (2 VALU ops/elem),
//     accumulation offloaded to the matrix pipe: chained V_WMMA_F32_16X16X4_F32 with A=ones
//   - output written once with nontemporal stores (streaming, don't pollute L2)

#define ROWS 4096
#define COLS 16384
#define TPB  256            // 8 waves of 32
#define V4PT 16             // float4 chunks per thread: 16*4 = 64 elems
#define NIT  50
#define BUDGETF 2.0f

typedef float v2f __attribute__((ext_vector_type(2)));
typedef float v4f __attribute__((ext_vector_type(4)));
typedef float v8f __attribute__((ext_vector_type(8)));

__device__ __forceinline__ float clip01(float x) {
  // clip(x, 0, 1) == median3(x, 0, 1)
  return __builtin_amdgcn_fmed3f(x, 0.0f, 1.0f);
}

__global__ __launch_bounds__(TPB) void budget_bisect_kernel(
    const float* __restrict__ X, float* __restrict__ Out) {
  // Single shared symbol -> assumed at LDS byte offset 0 (needed by async asm below).
  __shared__ float smem[COLS + 32];   // [0,COLS): staged row; [COLS,COLS+32): reduce scratch

  const int row  = blockIdx.x;
  const int t    = threadIdx.x;
  const int lane = t & 31;
  const int wave = t >> 5;

  const float* rowPtr = X + (size_t)row * COLS;     // uniform -> SGPR pair for saddr

  // ---- Stage the row into LDS via the CDNA5 async pipe (GVS addressing) ----
  // Each lane moves 16 B per issue; 16 issues/thread covers the 64 KB row.
  #pragma unroll
  for (int k = 0; k < V4PT; ++k) {
    unsigned idx4 = (unsigned)(k * TPB + t);        // lane-contiguous float4 index
    unsigned goff = idx4 * 16u;                     // global byte offset within row
    unsigned loff = idx4 * 16u;                     // LDS byte offset (smem at offset 0)
    asm volatile("global_load_async_to_lds_b128 %0, %1, %2"
                 :: "v"(loff), "v"(goff), "s"(rowPtr)
                 : "memory");
  }
  asm volatile("s_wait_asynccnt 0" ::: "memory");
  __syncthreads();

  // ---- Pull my 64 elements into VGPRs (conflict-free: lane-consecutive 16B) ----
  v4f xv[V4PT];
  #pragma unroll
  for (int k = 0; k < V4PT; ++k) {
    xv[k] = *reinterpret_cast<const v4f*>(&smem[(unsigned)(k * TPB + t) * 4u]);
  }

  // ---- Row max / min (once) ----
  float mx = -__builtin_inff();
  float mn =  __builtin_inff();
  #pragma unroll
  for (int k = 0; k < V4PT; ++k) {
    v4f x = xv[k];
    mx = fmaxf(mx, fmaxf(fmaxf(x.x, x.y), fmaxf(x.z, x.w)));
    mn = fminf(mn, fminf(fminf(x.x, x.y), fminf(x.z, x.w)));
  }
  #pragma unroll
  for (int m = 16; m >= 1; m >>= 1) {
    mx = fmaxf(mx, __shfl_xor(mx, m, 32));
    mn = fminf(mn, __shfl_xor(mn, m, 32));
  }
  if (lane == 0) { smem[COLS + wave] = mx; smem[COLS + 8 + wave] = mn; }
  __syncthreads();
  mx = smem[COLS + 0]; mn = smem[COLS + 8];
  #pragma unroll
  for (int w = 1; w < 8; ++w) {
    mx = fmaxf(mx, smem[COLS + w]);
    mn = fminf(mn, smem[COLS + 8 + w]);
  }

  const float bd = BUDGETF / (float)COLS;
  float tau_lo = mn - bd;                 // tau_hi - tau_lo == mx - mn
  float dm     = mx - mn;
  float tau_m  = tau_lo;

  const v2f ones = {1.0f, 1.0f};          // A matrix (16x4) of all ones -> D = sum_k(B) + C

  for (int it = 0; it < NIT; ++it) {
    dm *= 0.5f;
    tau_m = tau_lo + dm;

    v8f acc = {};                         // chained WMMA accumulator (C -> D)
    #pragma unroll
    for (int k = 0; k < V4PT; ++k) {
      v4f x = xv[k];
      v2f b0, b1;
      b0.x = clip01(x.x - tau_m);
      b0.y = clip01(x.y - tau_m);
      b1.x = clip01(x.z - tau_m);
      b1.y = clip01(x.w - tau_m);
      acc = __builtin_amdgcn_wmma_f32_16x16x4_f32(false, ones, false, b0,
                                                  (short)0, acc, false, false);
      acc = __builtin_amdgcn_wmma_f32_16x16x4_f32(false, ones, false, b1,
                                                  (short)0, acc, false, false);
    }
    // acc[0] lane n (n<16, duplicated in lanes 16..31) = column sum n.
    // Sum the 16 columns -> wave total (2048 elems) in every lane.
    float ws = acc[0];
    ws += __shfl_xor(ws, 1, 32);
    ws += __shfl_xor(ws, 2, 32);
    ws += __shfl_xor(ws, 4, 32);
    ws += __shfl_xor(ws, 8, 32);

    __syncthreads();                      // previous iteration's scratch readers done
    if (lane == 0) smem[COLS + 16 + wave] = ws;
    __syncthreads();
    float rowSum = 0.0f;
    #pragma unroll
    for (int w = 0; w < 8; ++w) rowSum += smem[COLS + 16 + w];

    float f = rowSum - BUDGETF;
    if (f >= 0.0f) tau_lo = tau_m;        // uniform across block (identical arithmetic)
  }

  // ---- p from the last midpoint; stream out once ----
  float* orow = Out + (size_t)row * COLS;
  #pragma unroll
  for (int k = 0; k < V4PT; ++k) {
    v4f x = xv[k], p;
    p.x = clip01(x.x - tau_m);
    p.y = clip01(x.y - tau_m);
    p.z = clip01(x.z - tau_m);
    p.w = clip01(x.w - tau_m);
    unsigned idx4 = (unsigned)(k * TPB + t);
    __builtin_nontemporal_store(p, reinterpret_cast<v4f*>(orow) + idx4);
  }
}

extern "C" void kernel_launch(void* const* d_in, const int* in_sizes, int n_in,
                              void* d_out, int out_size, void* d_ws, size_t ws_size,
                              hipStream_t stream) {
  (void)in_sizes; (void)n_in; (void)out_size; (void)d_ws; (void)ws_size;
  const float* X = (const float*)d_in[0];
  float* Out = (float*)d_out;
  hipLaunchKernelGGL(budget_bisect_kernel, dim3(ROWS), dim3(TPB), 0, stream, X, Out);
}